// LSTM_74131135529056
// MI455X (gfx1250) — compile-verified
//
#include <hip/hip_runtime.h>

#define SEQL 256
#define BATCH 64
#define IN_DIM 512
#define HID 1024

typedef __attribute__((ext_vector_type(2))) float v2f;
typedef __attribute__((ext_vector_type(8))) float v8f;

struct GateParams {
    const float* Wi[4];   // [HID][IN_DIM] row-major
    const float* Wh[4];   // [HID][HID]    row-major
};

__device__ __forceinline__ float fast_sigmoid(float x) {
    return 1.0f / (1.0f + __expf(-x));
}

// Dual-accumulator K loop: even K-steps -> acc0, odd -> acc1 (two independent
// WMMA dependency chains, 16 outstanding b64 loads per unrolled iteration).
__device__ __forceinline__ void kloop_dual(
    const float* __restrict__ a_base, const float* __restrict__ b_base,
    int kbegin, int kend, v8f& acc0, v8f& acc1)
{
    #pragma unroll 4
    for (int k0 = kbegin; k0 < kend; k0 += 8) {
        v2f a0 = *(const v2f*)(a_base + k0);
        v2f b0 = *(const v2f*)(b_base + k0);
        v2f a1 = *(const v2f*)(a_base + k0 + 4);
        v2f b1 = *(const v2f*)(b_base + k0 + 4);
        acc0 = __builtin_amdgcn_wmma_f32_16x16x4_f32(
            false, a0, false, b0, (short)0, acc0, false, false);
        acc1 = __builtin_amdgcn_wmma_f32_16x16x4_f32(
            false, a1, false, b1, (short)0, acc1, false, false);
    }
}

// One LSTM timestep, fused x-proj + h-proj + gates.
// grid = (BATCH/16, HID/16), block = 256 threads = 8 waves.
// wave w: gate g = w&3, split-K half = w>>2 (768 K each: balanced).
__global__ __launch_bounds__(256) void lstm_step_kernel(
    const float* __restrict__ x_t,      // [BATCH][IN_DIM] slice for step t
    const float* __restrict__ h_in,     // [BATCH][HID]
    float* __restrict__ h_out,          // [BATCH][HID]
    float* __restrict__ c_state,        // [BATCH][HID] (read-modify-write)
    float* __restrict__ h_seq_t,        // d_out + t*BATCH*HID
    const float* __restrict__ bsum,     // [4][HID] precomputed b_i + b_h
    GateParams gp)
{
    __shared__ float smem[8 * 256];     // 8 partial 16x16 tiles

    const int tid  = threadIdx.x;
    const int wave = tid >> 5;
    const int lane = tid & 31;
    const int gate = wave & 3;
    const int half = wave >> 2;

    const int bt = blockIdx.x;          // batch tile  (0..3)
    const int ht = blockIdx.y;          // hidden tile (0..63)

    const int m    = lane & 15;         // row within 16-wide tile
    const int kk   = (lane >> 4) << 1;  // lanes 16-31 handle K pair {2,3}

    const int arow = bt * 16 + m;       // batch index for A loads
    const int wrow = ht * 16 + m;       // weight row (= output column) for B loads

    v8f acc0 = {};
    v8f acc1 = {};

    const float* a_h = h_in + arow * HID + kk;
    const float* b_h = gp.Wh[gate] + wrow * HID + kk;

    if (half == 0) {
        // input projection (K = 512) + recurrent K in [0, 256)
        kloop_dual(x_t + arow * IN_DIM + kk, gp.Wi[gate] + wrow * IN_DIM + kk,
                   0, IN_DIM, acc0, acc1);
        kloop_dual(a_h, b_h, 0, 256, acc0, acc1);
    } else {
        // recurrent K in [256, 1024)
        kloop_dual(a_h, b_h, 256, HID, acc0, acc1);
    }

    // Spill accumulator tile to LDS.
    // C/D layout: VGPR r, lanes 0-15 -> M=r, N=lane; lanes 16-31 -> M=r+8, N=lane-16
    {
        float* s = smem + wave * 256;
        const int n     = lane & 15;
        const int mbase = (lane >> 4) << 3;   // 0 or 8
        #pragma unroll
        for (int r = 0; r < 8; ++r)
            s[(mbase + r) * 16 + n] = acc0[r] + acc1[r];
    }
    __syncthreads();

    // Elementwise gates: 256 threads, one tile element each.
    {
        const int e  = tid;            // 0..255 -> [mm][n]
        const int mm = e >> 4;
        const int n  = e & 15;
        const int b  = bt * 16 + mm;
        const int h  = ht * 16 + n;

        float fpre = smem[0 * 256 + e] + smem[4 * 256 + e] + bsum[0 * HID + h];
        float ipre = smem[1 * 256 + e] + smem[5 * 256 + e] + bsum[1 * HID + h];
        float opre = smem[2 * 256 + e] + smem[6 * 256 + e] + bsum[2 * HID + h];
        float gpre = smem[3 * 256 + e] + smem[7 * 256 + e] + bsum[3 * HID + h];

        float f = fast_sigmoid(fpre);
        float i = fast_sigmoid(ipre);
        float o = fast_sigmoid(opre);
        float g = tanhf(gpre);

        const int idx = b * HID + h;
        float c_new = f * c_state[idx] + i * g;
        c_state[idx] = c_new;
        float h_new = o * tanhf(c_new);
        h_out[idx]   = h_new;
        h_seq_t[idx] = h_new;
    }
}

// Zero h0/c0 and precompute per-gate bias sums (b_i + b_h).
__global__ __launch_bounds__(256) void prep_kernel(
    float* __restrict__ h0, float* __restrict__ c0,
    float* __restrict__ bsum,
    const float* __restrict__ bi0, const float* __restrict__ bh0,
    const float* __restrict__ bi1, const float* __restrict__ bh1,
    const float* __restrict__ bi2, const float* __restrict__ bh2,
    const float* __restrict__ bi3, const float* __restrict__ bh3)
{
    int i = blockIdx.x * blockDim.x + threadIdx.x;
    if (i < BATCH * HID) {
        h0[i] = 0.0f;
        c0[i] = 0.0f;
    }
    if (i < HID) {
        bsum[0 * HID + i] = bi0[i] + bh0[i];
        bsum[1 * HID + i] = bi1[i] + bh1[i];
        bsum[2 * HID + i] = bi2[i] + bh2[i];
        bsum[3 * HID + i] = bi3[i] + bh3[i];
    }
}

__global__ __launch_bounds__(256) void tail_copy_kernel(
    const float* __restrict__ h_fin, const float* __restrict__ c_fin,
    float* __restrict__ out_tail)
{
    int i = blockIdx.x * blockDim.x + threadIdx.x;
    if (i < BATCH * HID) {
        out_tail[i]               = h_fin[i];
        out_tail[BATCH * HID + i] = c_fin[i];
    }
}

extern "C" void kernel_launch(void* const* d_in, const int* in_sizes, int n_in,
                              void* d_out, int out_size, void* d_ws, size_t ws_size,
                              hipStream_t stream) {
    (void)in_sizes; (void)n_in; (void)out_size; (void)ws_size;

    const float* x = (const float*)d_in[0];
    GateParams gp;
    const float* bi[4];
    const float* bh[4];
    // setup_inputs order: x, then per gate g in [f,i,o,c]: W_ig, b_ig, W_hg, b_hg
    for (int g = 0; g < 4; ++g) {
        gp.Wi[g] = (const float*)d_in[1 + 4 * g];
        bi[g]    = (const float*)d_in[2 + 4 * g];
        gp.Wh[g] = (const float*)d_in[3 + 4 * g];
        bh[g]    = (const float*)d_in[4 + 4 * g];
    }

    float* ws    = (float*)d_ws;
    float* hbuf0 = ws;                    // BATCH*HID
    float* hbuf1 = ws + BATCH * HID;      // BATCH*HID
    float* cbuf  = ws + 2 * BATCH * HID;  // BATCH*HID
    float* bsum  = ws + 3 * BATCH * HID;  // 4*HID
    float* out   = (float*)d_out;

    // h0 = 0, c0 = 0, bias sums (must re-init every call)
    prep_kernel<<<(BATCH * HID + 255) / 256, 256, 0, stream>>>(
        hbuf0, cbuf, bsum,
        bi[0], bh[0], bi[1], bh[1], bi[2], bh[2], bi[3], bh[3]);

    dim3 grid(BATCH / 16, HID / 16);
    for (int t = 0; t < SEQL; ++t) {
        const float* h_in  = (t & 1) ? hbuf1 : hbuf0;
        float*       h_out = (t & 1) ? hbuf0 : hbuf1;
        lstm_step_kernel<<<grid, 256, 0, stream>>>(
            x + (size_t)t * BATCH * IN_DIM, h_in, h_out, cbuf,
            out + (size_t)t * BATCH * HID, bsum, gp);
    }
    // After t=255 (odd), final h is in hbuf0.
    tail_copy_kernel<<<(BATCH * HID + 255) / 256, 256, 0, stream>>>(
        hbuf0, cbuf, out + (size_t)SEQL * BATCH * HID);
}